// GraphTransformerLayer_75453985456264
// MI455X (gfx1250) — compile-verified
//
#include <hip/hip_runtime.h>
#include <hip/hip_bf16.h>

// ---------------------------------------------------------------------------
// GraphTransformerLayer for MI455X (gfx1250).
// N=50000, DIM=128, H=8, HD=16, E=800000.
// GEMMs on V_WMMA_F32_16X16X4_F32 (f32 in/out: edge math is the bottleneck,
// whole working set is L2-resident, so fp32 fidelity is free).
// ---------------------------------------------------------------------------

typedef __attribute__((ext_vector_type(2))) float v2f;
typedef __attribute__((ext_vector_type(8))) float v8f;

#define DIMC 128
#define EPS 1e-5f

// ---- monotone float<->uint encoding for atomicMax on floats ---------------
__device__ __forceinline__ unsigned fenc(float f) {
  unsigned u = __float_as_uint(f);
  return (u & 0x80000000u) ? ~u : (u | 0x80000000u);
}
__device__ __forceinline__ float fdec(unsigned u) {
  u = (u & 0x80000000u) ? (u & 0x7FFFFFFFu) : ~u;
  return __uint_as_float(u);
}

// ---------------------------------------------------------------------------
// BatchNorm column statistics (deterministic two-pass tree reduction).
// ---------------------------------------------------------------------------
__global__ void bn_stats_partial(const float* __restrict__ X, float* __restrict__ part, int M) {
  const int c = threadIdx.x;  // 128 threads: one column each (coalesced rows)
  float s = 0.f, q = 0.f;
  for (int r = blockIdx.x; r < M; r += gridDim.x) {
    float v = X[(size_t)r * DIMC + c];
    s += v;
    q += v * v;
  }
  part[(size_t)blockIdx.x * 256 + c] = s;
  part[(size_t)blockIdx.x * 256 + 128 + c] = q;
}

__global__ void bn_stats_reduce(const float* __restrict__ part, int nblk,
                                const float* __restrict__ gam, const float* __restrict__ bet,
                                float* __restrict__ ss, int M) {
  const int c = threadIdx.x;  // 128
  float s = 0.f, q = 0.f;
  for (int i = 0; i < nblk; ++i) {
    s += part[(size_t)i * 256 + c];
    q += part[(size_t)i * 256 + 128 + c];
  }
  const float mean = s / (float)M;
  const float var = q / (float)M - mean * mean;
  const float inv = rsqrtf(var + EPS);
  const float sc = gam[c] * inv;
  ss[c] = sc;                      // scale
  ss[128 + c] = bet[c] - mean * sc;  // shift
}

// ---------------------------------------------------------------------------
// Zero / -inf initialization of accumulation buffers.
// ---------------------------------------------------------------------------
__global__ void init_buffers(float* __restrict__ asum, unsigned* __restrict__ smax,
                             float* __restrict__ wV, int n8, int n128) {
  const int i = blockIdx.x * blockDim.x + threadIdx.x;
  if (i < n8) {
    asum[i] = 0.f;
    smax[i] = fenc(-1e30f);
  }
  if (i < n128) wV[i] = 0.f;
}

// ---------------------------------------------------------------------------
// Templated WMMA f32 GEMM:  out[M,NW] = epi( A[M,KDIM] @ W[KDIM,NW] )
//   NORM_IN : A -> A*ss[k] + ss[128+k]   (fused BatchNorm)
//   ADD_BIAS: += bias[n]
//   RELU_OUT: max(0,.)
//   ADD_RES : += res[m,n]
// Block = 256 threads (8 waves). Each wave: 16 rows x 128 cols (8 C-tiles).
// W staged through LDS in 64-row K chunks.
// ---------------------------------------------------------------------------
template <int KDIM, bool NORM_IN, bool ADD_BIAS, bool RELU_OUT, bool ADD_RES>
__global__ __launch_bounds__(256) void wmma_gemm(
    const float* __restrict__ A, const float* __restrict__ W, int NW,
    const float* __restrict__ bias, const float* __restrict__ ss,
    const float* __restrict__ res, float* __restrict__ out, int M) {
  __shared__ float lw[64 * 132];  // 64 K-rows x 128 cols, padded stride 132

  const int tid = threadIdx.x;
  const int wave = tid >> 5;
  const int lane = tid & 31;
  const int half = lane >> 4;   // 0: lanes 0-15, 1: lanes 16-31
  const int lrow = lane & 15;
  const int r0 = blockIdx.x * 128 + wave * 16;
  const int ncol0 = blockIdx.y * 128;

  const v8f vzero = {0.f, 0.f, 0.f, 0.f, 0.f, 0.f, 0.f, 0.f};
  v8f acc[8];
#pragma unroll
  for (int t = 0; t < 8; ++t) acc[t] = vzero;

  int arow = r0 + lrow;
  if (arow >= M) arow = M - 1;  // clamp; clamped rows are never stored
  const float* Arow = A + (size_t)arow * KDIM;

  for (int kc = 0; kc < KDIM; kc += 64) {
    // stage W chunk into LDS
    for (int i = tid; i < 64 * 128; i += 256) {
      const int kk = i >> 7, nn = i & 127;
      lw[kk * 132 + nn] = W[(size_t)(kc + kk) * NW + ncol0 + nn];
    }
    __syncthreads();

#pragma unroll
    for (int k0 = 0; k0 < 64; k0 += 4) {
      const int klocal = k0 + half * 2;      // per-lane K pair within chunk
      const int kk = kc + klocal;            // global K
      float a0 = Arow[kk];
      float a1 = Arow[kk + 1];
      if constexpr (NORM_IN) {
        a0 = a0 * ss[kk] + ss[128 + kk];
        a1 = a1 * ss[kk + 1] + ss[128 + kk + 1];
      }
      v2f av;
      av.x = a0;
      av.y = a1;
#pragma unroll
      for (int t = 0; t < 8; ++t) {
        v2f bv;
        bv.x = lw[klocal * 132 + t * 16 + lrow];
        bv.y = lw[(klocal + 1) * 132 + t * 16 + lrow];
        acc[t] = __builtin_amdgcn_wmma_f32_16x16x4_f32(
            false, av, false, bv, (short)0, acc[t], false, false);
      }
    }
    __syncthreads();
  }

  // epilogue: C tile layout -> VGPR i holds (m = r0 + half*8 + i, n = ncol0 + t*16 + lrow)
#pragma unroll
  for (int t = 0; t < 8; ++t) {
    const int n = ncol0 + t * 16 + lrow;
#pragma unroll
    for (int i = 0; i < 8; ++i) {
      const int m = r0 + half * 8 + i;
      if (m < M) {
        float v = acc[t][i];
        if constexpr (ADD_BIAS) v += bias[n];
        if constexpr (RELU_OUT) v = fmaxf(v, 0.f);
        if constexpr (ADD_RES) v += res[(size_t)m * NW + n];
        out[(size_t)m * NW + n] = v;
      }
    }
  }
}

// ---------------------------------------------------------------------------
// Edge kernels: one thread per (edge, head). Q/K/V are L2-resident (25.6 MB ea).
// ---------------------------------------------------------------------------
__global__ void edge_score(const float* __restrict__ Qm, const float* __restrict__ Km,
                           const int* __restrict__ src, const int* __restrict__ dst,
                           float* __restrict__ score, unsigned* __restrict__ smax, int EH) {
  const int idx = blockIdx.x * blockDim.x + threadIdx.x;
  if (idx >= EH) return;
  const int e = idx >> 3, hh = idx & 7;
  const int s = src[e], d = dst[e];
  const float4* kp = (const float4*)(Km + (size_t)s * DIMC + hh * 16);
  const float4* qp = (const float4*)(Qm + (size_t)d * DIMC + hh * 16);
  float acc = 0.f;
#pragma unroll
  for (int i = 0; i < 4; ++i) {
    const float4 kv = kp[i];
    const float4 qv = qp[i];
    acc += kv.x * qv.x + kv.y * qv.y + kv.z * qv.z + kv.w * qv.w;
  }
  acc *= 0.25f;  // 1/sqrt(HD)
  acc = fminf(5.f, fmaxf(-5.f, acc));
  score[idx] = acc;
  atomicMax(&smax[d * 8 + hh], fenc(acc));
}

__global__ void edge_exp(const int* __restrict__ dst, float* __restrict__ score,
                         const unsigned* __restrict__ smax, float* __restrict__ asum, int EH) {
  const int idx = blockIdx.x * blockDim.x + threadIdx.x;
  if (idx >= EH) return;
  const int e = idx >> 3, hh = idx & 7;
  const int d = dst[e];
  const float a = __expf(score[idx] - fdec(smax[d * 8 + hh]));
  score[idx] = a;
  atomicAdd(&asum[d * 8 + hh], a);
}

__global__ void edge_aggregate(const float* __restrict__ Vm, const int* __restrict__ src,
                               const int* __restrict__ dst, const float* __restrict__ a,
                               const float* __restrict__ asum, float* __restrict__ wV, int EH) {
  const int idx = blockIdx.x * blockDim.x + threadIdx.x;
  if (idx >= EH) return;
  const int e = idx >> 3, hh = idx & 7;
  const int s = src[e], d = dst[e];
  const float w = a[idx] / asum[d * 8 + hh];
  const float4* vp = (const float4*)(Vm + (size_t)s * DIMC + hh * 16);
  float* op = wV + (size_t)d * DIMC + hh * 16;
#pragma unroll
  for (int i = 0; i < 4; ++i) {
    const float4 vv = vp[i];
    atomicAdd(op + i * 4 + 0, vv.x * w);
    atomicAdd(op + i * 4 + 1, vv.y * w);
    atomicAdd(op + i * 4 + 2, vv.z * w);
    atomicAdd(op + i * 4 + 3, vv.w * w);
  }
}

// ---------------------------------------------------------------------------
// Launch orchestration.
// ---------------------------------------------------------------------------
extern "C" void kernel_launch(void* const* d_in, const int* in_sizes, int n_in,
                              void* d_out, int out_size, void* d_ws, size_t ws_size,
                              hipStream_t stream) {
  (void)n_in; (void)out_size; (void)ws_size;
  const float* h   = (const float*)d_in[0];
  const int*   src = (const int*)d_in[1];
  const int*   dst = (const int*)d_in[2];
  const float* Wq  = (const float*)d_in[3];
  const float* Wk  = (const float*)d_in[4];
  const float* Wv  = (const float*)d_in[5];
  const float* Wo  = (const float*)d_in[6];
  const float* bo  = (const float*)d_in[7];
  const float* g1  = (const float*)d_in[8];
  const float* be1 = (const float*)d_in[9];
  const float* g2  = (const float*)d_in[10];
  const float* be2 = (const float*)d_in[11];
  const float* W1  = (const float*)d_in[12];
  const float* bb1 = (const float*)d_in[13];
  const float* W2  = (const float*)d_in[14];
  const float* bb2 = (const float*)d_in[15];

  const int Nn = in_sizes[0] / DIMC;
  const int E  = in_sizes[1];
  const int EH = E * 8;

  float* ws = (float*)d_ws;
  size_t o = 0;
  float* Qm    = ws + o; o += (size_t)Nn * DIMC;
  float* Km    = ws + o; o += (size_t)Nn * DIMC;
  float* Vm    = ws + o; o += (size_t)Nn * DIMC;
  float* score = ws + o; o += (size_t)EH;
  unsigned* smax = (unsigned*)(ws + o); o += (size_t)Nn * 8;
  float* asum  = ws + o; o += (size_t)Nn * 8;
  float* wV    = ws + o; o += (size_t)Nn * DIMC;
  float* h2    = ws + o; o += (size_t)Nn * DIMC;
  float* part  = ws + o; o += (size_t)512 * 256;
  float* ss1   = ws + o; o += 256;
  float* ss2   = ws + o; o += 256;
  float* T     = Qm;  // FFN intermediate [Nn,256] aliases dead Q+K region

  const dim3 blk256(256);
  const dim3 gM((Nn + 127) / 128, 1);
  const dim3 gM2((Nn + 127) / 128, 2);
  const int eb = (EH + 255) / 256;

  // BN1 statistics
  bn_stats_partial<<<512, 128, 0, stream>>>(h, part, Nn);
  bn_stats_reduce<<<1, 128, 0, stream>>>(part, 512, g1, be1, ss1, Nn);

  // init accumulators
  init_buffers<<<((size_t)Nn * DIMC + 255) / 256, blk256, 0, stream>>>(asum, smax, wV, Nn * 8, Nn * DIMC);

  // Q/K/V = bn1(h) @ W{q,k,v}
  wmma_gemm<128, true, false, false, false><<<gM, blk256, 0, stream>>>(h, Wq, 128, nullptr, ss1, nullptr, Qm, Nn);
  wmma_gemm<128, true, false, false, false><<<gM, blk256, 0, stream>>>(h, Wk, 128, nullptr, ss1, nullptr, Km, Nn);
  wmma_gemm<128, true, false, false, false><<<gM, blk256, 0, stream>>>(h, Wv, 128, nullptr, ss1, nullptr, Vm, Nn);

  // edge softmax + aggregation
  edge_score<<<eb, blk256, 0, stream>>>(Qm, Km, src, dst, score, smax, EH);
  edge_exp<<<eb, blk256, 0, stream>>>(dst, score, smax, asum, EH);
  edge_aggregate<<<eb, blk256, 0, stream>>>(Vm, src, dst, score, asum, wV, EH);

  // h2 = h + wV @ Wo + bo
  wmma_gemm<128, false, true, false, true><<<gM, blk256, 0, stream>>>(wV, Wo, 128, bo, nullptr, h, h2, Nn);

  // BN2 statistics
  bn_stats_partial<<<512, 128, 0, stream>>>(h2, part, Nn);
  bn_stats_reduce<<<1, 128, 0, stream>>>(part, 512, g2, be2, ss2, Nn);

  // T = relu(bn2(h2) @ W1 + b1)   [Nn, 256]
  wmma_gemm<128, true, true, true, false><<<gM2, blk256, 0, stream>>>(h2, W1, 256, bb1, ss2, nullptr, T, Nn);

  // out = h2 + T @ W2 + b2
  wmma_gemm<256, false, true, false, true><<<gM, blk256, 0, stream>>>(T, W2, 128, bb2, nullptr, h2, (float*)d_out, Nn);
}